// FFF_74062416053414
// MI455X (gfx1250) — compile-verified
//
#include <hip/hip_runtime.h>
#include <hip/hip_bf16.h>

typedef __attribute__((ext_vector_type(2))) float v2f;
typedef __attribute__((ext_vector_type(8))) float v8f;

#if defined(__HIP_DEVICE_COMPILE__) && !__has_builtin(__builtin_amdgcn_wmma_f32_16x16x4_f32)
#error "gfx1250: __builtin_amdgcn_wmma_f32_16x16x4_f32 not available"
#endif

#define FFF_DEPTH 10
#define NIN  1024
#define NOUT 1024

// One block = 16 samples. 8 wave32 waves:
//   - each wave owns a 128-wide K-slice of nIn for the lam dot (WMMA diag trick)
//   - each wave owns a 128-wide slice of nOut, y kept in 8 resident C-fragments
__global__ __launch_bounds__(256)
void FFF_74062416053414_kernel(const float* __restrict__ x,
                               const float* __restrict__ w1s,
                               const float* __restrict__ w2s,
                               float* __restrict__ y)
{
    const int tid  = threadIdx.x;
    const int lane = tid & 31;
    const int wave = tid >> 5;             // 0..7
    const int r0   = blockIdx.x * 16;      // sample-tile base row

    __shared__ int   nodeS[16];
    __shared__ float lamS[16];
    __shared__ float lamPart[8][16];

    if (tid < 16) nodeS[tid] = 0;

    const int m    = lane & 15;            // sample index within tile (row of A, col of B)
    const int koff = (lane >> 4) << 1;     // 0 for lanes 0-15, 2 for lanes 16-31

    // ---- Preload A fragments: x rows r0..r0+15 over this wave's K-slice.
    // Read once from HBM, reused across all 10 levels (64 VGPRs).
    const int kbase = wave * 128;
    v2f afrag[32];
#pragma unroll
    for (int kk = 0; kk < 32; ++kk) {
        const float* p = x + (size_t)(r0 + m) * NIN + kbase + kk * 4 + koff;
        afrag[kk] = *(const v2f*)p;
    }

    // ---- y accumulators: 8 tiles of 16x16 f32 (cols wave*128 + t*16 .. +15)
    v8f yacc[8];
#pragma unroll
    for (int t = 0; t < 8; ++t) yacc[t] = (v8f){0.f,0.f,0.f,0.f,0.f,0.f,0.f,0.f};

    for (int d = 0; d < FFF_DEPTH; ++d) {
        __syncthreads();  // nodeS valid

        // ===== Phase 1: lam partials = diag( X_tile * W1g^T ) over K-slice =====
        // B[k][n] = w1s[node_n][kbase+k]; lane supplies column n = m.
        const int myNode = nodeS[m];
        const float* w1row = w1s + (size_t)myNode * NIN + kbase + koff;
        v8f c = (v8f){0.f,0.f,0.f,0.f,0.f,0.f,0.f,0.f};
#pragma unroll
        for (int kk = 0; kk < 32; ++kk) {
            v2f b = *(const v2f*)(w1row + kk * 4);
            c = __builtin_amdgcn_wmma_f32_16x16x4_f32(
                    false, afrag[kk], false, b, (short)0, c, false, false);
        }
        // Extract diagonal: diag[v] -> lane v, VGPR v; diag[v+8] -> lane v+24, VGPR v.
#pragma unroll
        for (int v = 0; v < 8; ++v) {
            if (lane == v)      lamPart[wave][v]     = c[v];
            if (lane == v + 24) lamPart[wave][v + 8] = c[v];
        }
        __syncthreads();

        // ===== Phase 2: reduce 8 wave partials -> lam =====
        if (tid < 16) {
            float s = 0.f;
#pragma unroll
            for (int w = 0; w < 8; ++w) s += lamPart[w][tid];
            lamS[tid] = s;
        }
        __syncthreads();

        // ===== Phase 3: Y_tile += diag(lam) * W2g over this wave's 128 cols =====
        const float lm = lamS[m];
#pragma unroll
        for (int t = 0; t < 8; ++t) {
            const int c0 = wave * 128 + t * 16;
#pragma unroll
            for (int j = 0; j < 4; ++j) {
                const int k0 = 4 * j + koff;
                v2f a = { (m == k0    ) ? lm : 0.f,
                          (m == k0 + 1) ? lm : 0.f };
                // coalesced: 16 lanes read 64B contiguous per row
                v2f b = { w2s[(size_t)nodeS[k0]     * NOUT + c0 + m],
                          w2s[(size_t)nodeS[k0 + 1] * NOUT + c0 + m] };
                yacc[t] = __builtin_amdgcn_wmma_f32_16x16x4_f32(
                              false, a, false, b, (short)0, yacc[t], false, false);
            }
        }
        __syncthreads();

        // ===== Phase 4: descend tree; prefetch next level's w1 row =====
        if (tid < 16) {
            const int nn = nodeS[tid] * 2 + 1 + (lamS[tid] > 0.f ? 1 : 0);
            nodeS[tid] = nn;
            if (d + 1 < FFF_DEPTH) {
                __builtin_prefetch(w1s + (size_t)nn * NIN, 0, 1);   // global_prefetch_b8
            }
        }
    }

    // ---- Write y once: element (M,N) of tile t: row r0+v+(lane>=16?8:0), col c0+m
    const int mrow = (lane >> 4) ? 8 : 0;
#pragma unroll
    for (int t = 0; t < 8; ++t) {
        const int c0 = wave * 128 + t * 16;
#pragma unroll
        for (int v = 0; v < 8; ++v) {
            y[(size_t)(r0 + v + mrow) * NOUT + c0 + m] = yacc[t][v];
        }
    }
}

extern "C" void kernel_launch(void* const* d_in, const int* in_sizes, int n_in,
                              void* d_out, int out_size, void* d_ws, size_t ws_size,
                              hipStream_t stream) {
    const float* x   = (const float*)d_in[0];
    const float* w1s = (const float*)d_in[1];
    const float* w2s = (const float*)d_in[2];
    float* y = (float*)d_out;

    const int B = in_sizes[0] / NIN;       // 65536
    const int blocks = B / 16;             // 4096 sample tiles

    FFF_74062416053414_kernel<<<blocks, 256, 0, stream>>>(x, w1s, w2s, y);
}